// TorchSplitIFTFieldCore_12343736009066
// MI455X (gfx1250) — compile-verified
//
#include <hip/hip_runtime.h>
#include <hip/hip_bf16.h>
#include <math.h>

#define Hd 1536
#define Wd 1536
#define HW (Hd * Wd)
#define TX 32
#define TY 8
#define NBX (Wd / TX)          // 48
#define NBY (Hd / TY)          // 192
#define NB  (NBX * NBY)        // 9216 blocks
#define TPITCH 36              // TX+4, contiguous rows (TDM-friendly, no pad needed)
#define TROWS  12              // TY + 4 halo
#define TILE_ELEMS (TROWS * TPITCH)   // 432 floats per channel tile

typedef float v2f __attribute__((ext_vector_type(2)));
typedef float v8f __attribute__((ext_vector_type(8)));
typedef unsigned int u32x4 __attribute__((ext_vector_type(4)));
typedef int i32x4 __attribute__((ext_vector_type(4)));
typedef int i32x8 __attribute__((ext_vector_type(8)));

__device__ __forceinline__ int wrapi(int x, int n) {
    if (x < 0) x += n;
    else if (x >= n) x -= n;
    return x;
}

__device__ __forceinline__ float clip4(float x) {
    return fminf(fmaxf(x, -4.0f), 4.0f);
}

// One fused Euler step: TDM (interior) / wrap-loop (edge) halo staging to LDS,
// 13-pt stencil, WMMA channel mix, reactions, integrate+clip, block partials.
__global__ __launch_bounds__(256)
void ift_step_kernel(const float* __restrict__ phi,
                     const float* __restrict__ sens,
                     const float* __restrict__ gamma,
                     const float* __restrict__ alpha,
                     const float* __restrict__ mixw,
                     float* __restrict__ out,
                     float* __restrict__ partials) {
    // 12 channel slots: 10 real + 2 zero (K-padding for WMMA). Contiguous tiles.
    __shared__ __align__(16) float sT[12][TILE_ELEMS];   // 20.7 KB halo tiles
    __shared__ __align__(16) float sW[16 * 12];          // mix_w padded 16x12
    __shared__ __align__(16) float sPhiT[256 * 12];      // center phi, ch-major
    __shared__ __align__(16) float sC[16][264];          // coupling / reduction

    const int tid = threadIdx.x;
    const int tx  = tid & 31;
    const int ty  = tid >> 5;
    const int bx  = blockIdx.x;
    const int by  = blockIdx.y;
    const int gx  = bx * TX + tx;
    const int gy  = by * TY + ty;

    const bool interior = (bx >= 1) & (bx <= NBX - 2) & (by >= 1) & (by <= NBY - 2);

    // ---- issue TDM for interior blocks first (overlap with LDS init) ----
    if (interior) {
        if (tid == 0) {
            const unsigned long long gaddr =
                (unsigned long long)(const void*)phi +
                (unsigned long long)(((by * TY - 2) * Wd + (bx * TX - 2)) * 4);
            const unsigned int ldsoff = (unsigned int)(unsigned long long)(void*)&sT[0][0];

            // D# group 0: count=1, lds_addr, global_addr, type=2
            u32x4 g0;
            g0[0] = 1u;                                    // count=1, user mode
            g0[1] = ldsoff;                                // lds_addr [63:32]
            g0[2] = (unsigned int)gaddr;                   // global_addr lo
            g0[3] = ((unsigned int)(gaddr >> 32) & 0x01FFFFFFu) | (2u << 30); // hi + type

            // D# group 1: data_size=4B, iterate_enable, dims/strides/tile
            const unsigned int tdim0 = (unsigned int)Wd;   // row length bound
            const unsigned int tdim1 = 1u << 20;           // generous (no OOB on interior)
            i32x8 g1;
            g1[0] = (int)((2u << 16) | (1u << 19));        // data_size=2(4B), iterate=1
            g1[1] = (int)((tdim0 & 0xFFFFu) << 16);        // tensor_dim0 lo16
            g1[2] = (int)((tdim0 >> 16) | ((tdim1 & 0xFFFFu) << 16));
            g1[3] = (int)((tdim1 >> 16) | ((unsigned)TPITCH << 16)); // tile_dim0=36
            g1[4] = TROWS;                                 // tile_dim1=12, tile_dim2=0
            g1[5] = Wd;                                    // tensor_dim0_stride lo32
            g1[6] = 0;
            g1[7] = 0;

            // D# group 2 (iterate mode): per-channel increments, 10 iterations
            i32x4 g2;
            g2[0] = 0;                                     // tensor_dim2 (unused)
            g2[1] = TILE_ELEMS;                            // lds_addr_increment (elems)
            g2[2] = HW;                                    // global_addr_increment lo32
            g2[3] = (int)(9u << 16);                       // iterate_count=9 (10x)

            i32x4 g3 = {0, 0, 0, 0};
            i32x8 g4 = {0, 0, 0, 0, 0, 0, 0, 0};          // extra group (clang-23 form)

            __builtin_amdgcn_tensor_load_to_lds(g0, g1, g2, g3, g4, 0);
        }
    } else {
        // Edge blocks: manual halo staging with periodic wrap.
        #pragma unroll
        for (int c = 0; c < 10; ++c) {
            const float* pc = phi + (size_t)c * HW;
            for (int idx = tid; idx < TILE_ELEMS; idx += 256) {
                const int r   = idx / TPITCH;
                const int col = idx - r * TPITCH;
                const int gr  = wrapi(by * TY + r - 2, Hd);
                const int gc  = wrapi(bx * TX + col - 2, Wd);
                sT[c][idx] = pc[gr * Wd + gc];
            }
        }
    }

    // ---- zero-pad channels 10,11 and stage padded mix_w (no races) ----
    {
        float* z = &sT[10][0];
        for (int idx = tid; idx < 2 * TILE_ELEMS; idx += 256) z[idx] = 0.0f;
    }
    if (tid < 192) {
        const int o = tid / 12, c = tid - o * 12;
        sW[tid] = (o < 10 && c < 10) ? mixw[o * 10 + c] : 0.0f;
    }

    if (interior && tid == 0) __builtin_amdgcn_s_wait_tensorcnt(0);
    __syncthreads();

    // ---- 13-point stencil: lap and bilap per channel ----
    float p_[10], lap_[10], bil_[10];
    const int base = (ty + 2) * TPITCH + (tx + 2);
    #pragma unroll
    for (int c = 0; c < 10; ++c) {
        const float* T = &sT[c][0];
        const float C  = T[base];
        const float a1 = T[base - TPITCH] + T[base + TPITCH] + T[base - 1] + T[base + 1];
        const float a2 = T[base - 2 * TPITCH] + T[base + 2 * TPITCH] + T[base - 2] + T[base + 2];
        const float dg = T[base - TPITCH - 1] + T[base - TPITCH + 1] +
                         T[base + TPITCH - 1] + T[base + TPITCH + 1];
        p_[c]   = C;
        lap_[c] = a1 - 4.0f * C;
        bil_[c] = a2 + 2.0f * dg - 8.0f * a1 + 20.0f * C;   // lap(lap)
    }

    // ---- stage center phi channel-major for WMMA B (zero K-pad) ----
    {
        float* dst = &sPhiT[tid * 12];
        #pragma unroll
        for (int c = 0; c < 10; ++c) dst[c] = p_[c];
        dst[10] = 0.0f; dst[11] = 0.0f;
    }
    __syncthreads();

    // ---- coupling = mix_w @ phi via V_WMMA_F32_16X16X4_F32 (K=12 in 3 steps) ----
    {
        const int lane   = tid & 31;
        const int waveId = tid >> 5;
        const int n      = lane & 15;          // M for A, N for B/D
        const int khalf  = lane >> 4;          // 0 -> K{0,1}, 1 -> K{2,3}
        #pragma unroll
        for (int h = 0; h < 2; ++h) {
            v8f acc = {};
            const int pix = waveId * 32 + h * 16 + n;
            #pragma unroll
            for (int ks = 0; ks < 3; ++ks) {
                const int k0 = ks * 4 + khalf * 2;
                const v2f a = *(const v2f*)&sW[n * 12 + k0];
                const v2f b = *(const v2f*)&sPhiT[pix * 12 + k0];
                acc = __builtin_amdgcn_wmma_f32_16x16x4_f32(
                          false, a, false, b, (short)0, acc, false, false);
            }
            #pragma unroll
            for (int r = 0; r < 8; ++r)
                sC[r + khalf * 8][pix] = acc[r];   // rows 10..15 dead
        }
    }
    __syncthreads();

    float cpl[10];
    #pragma unroll
    for (int c = 0; c < 10; ++c) cpl[c] = sC[c][tid];

    // ---- per-channel reactions ----
    float g[10];
    #pragma unroll
    for (int c = 0; c < 10; ++c) g[c] = gamma[c];
    const float a0 = alpha[0];

    float ch[10];
    {
        float p, p2, p3;
        p = p_[0]; ch[0] = -g[0] * lap_[0] - a0 * p * p * p + 0.12f * bil_[0];
        p = p_[1]; ch[1] = -g[1] * lap_[1] + 0.15f * __sinf(p) + 0.08f * p * p * p;
        p = p_[2]; ch[2] = -g[2] * lap_[2] + 0.2f * p * (1.0f - p * p);
        p = p_[3]; p2 = p * p;
                   ch[3] = -g[3] * lap_[3] - 0.03f * p2 * p2 * p + 0.08f * bil_[3];
        p = p_[4]; ch[4] = -g[4] * lap_[4] + 0.12f * p * __logf(fabsf(p) + 1e-6f)
                           + 0.02f * bil_[4];
        p = p_[5]; p2 = p * p;
                   ch[5] = -g[5] * lap_[5] + 0.08f * p2 * p - 0.02f * p2 * p2 * p;
        p = p_[6]; p2 = p * p;
                   ch[6] = -g[6] * lap_[6] + 0.06f * p2 * p2 - 0.04f * p;
        p = fminf(fmaxf(p_[7], -2.0f), 2.0f);
        { const float e = __expf(p);
                   ch[7] = -g[7] * lap_[7] + 0.04f * (e - 1.0f / e); } // 0.08*sinh
        p = p_[8]; ch[8] = -g[8] * lap_[8] + 0.05f * p * p - 0.08f * p;
        p = p_[9]; p2 = p * p; p3 = p2 * p;
                   ch[9] = -g[9] * lap_[9] + 0.02f * p3 * p3 - 0.04f * p3
                           + 0.01f * bil_[9];
    }

    const float fast_mean = 0.2f * (p_[0] + p_[1] + p_[2] + p_[3] + p_[4]);
    const float slow_mean = 0.2f * (p_[5] + p_[6] + p_[7] + p_[8] + p_[9]);
    const float sd = sens[gy * Wd + gx];

    float outv[10];
    #pragma unroll
    for (int c = 0; c < 5; ++c) {
        float d = ch[c] + 0.06f * cpl[c] + 0.02f * slow_mean;     // EPS_FAST, S2F
        if (c == 0) d += 0.2f * sd;
        if (c == 4) d += 0.08f * fabsf(sd);
        outv[c] = clip4(p_[c] + 0.02f * d);                       // DT * FAST_DT
    }
    #pragma unroll
    for (int c = 5; c < 10; ++c) {
        float d = ch[c] + 0.02f * cpl[c] + 0.04f * fast_mean;     // EPS_SLOW, F2S
        if (c == 9) d += 0.05f * sd;
        outv[c] = clip4(p_[c] + 0.004f * d);                      // DT * SLOW_DT
    }

    // ---- store phi_next (coalesced per channel) ----
    const int pix = gy * Wd + gx;
    #pragma unroll
    for (int c = 0; c < 10; ++c) out[(size_t)c * HW + pix] = outv[c];

    // ---- deterministic per-block partial sums (fixed tree order) ----
    #pragma unroll
    for (int c = 0; c < 10; ++c) sC[c][tid] = outv[c];
    __syncthreads();
    for (int s = 128; s > 0; s >>= 1) {
        if (tid < s) {
            #pragma unroll
            for (int c = 0; c < 10; ++c) sC[c][tid] += sC[c][tid + s];
        }
        __syncthreads();
    }
    if (tid < 10) {
        const int blk = by * NBX + bx;
        partials[tid * NB + blk] = sC[tid][0];
    }
}

// Final reduction: 10 blocks, one channel each; fixed order -> deterministic.
__global__ __launch_bounds__(256)
void ift_reduce_kernel(const float* __restrict__ partials,
                       float* __restrict__ out) {
    __shared__ float red[256];
    const int c = blockIdx.x;
    float s = 0.0f;
    for (int i = threadIdx.x; i < NB; i += 256) s += partials[c * NB + i];
    red[threadIdx.x] = s;
    __syncthreads();
    for (int k = 128; k > 0; k >>= 1) {
        if (threadIdx.x < k) red[threadIdx.x] += red[threadIdx.x + k];
        __syncthreads();
    }
    if (threadIdx.x == 0) {
        const float mean = red[0] * (1.0f / ((float)Hd * (float)Wd));
        out[(size_t)10 * HW + 2 + c] = mean;           // field_means[c]
        if (c == 0) out[(size_t)10 * HW]     = mean;   // phi1_mean
        if (c == 4) out[(size_t)10 * HW + 1] = mean;   // phi5_mean
    }
}

extern "C" void kernel_launch(void* const* d_in, const int* in_sizes, int n_in,
                              void* d_out, int out_size, void* d_ws, size_t ws_size,
                              hipStream_t stream) {
    (void)in_sizes; (void)n_in; (void)out_size; (void)ws_size;
    const float* phi   = (const float*)d_in[0];
    const float* sens  = (const float*)d_in[1];
    const float* gamma = (const float*)d_in[2];
    const float* alpha = (const float*)d_in[3];
    const float* mixw  = (const float*)d_in[4];
    float* out      = (float*)d_out;
    float* partials = (float*)d_ws;   // 10 * 9216 floats, fully rewritten each call

    dim3 grid(NBX, NBY);
    ift_step_kernel<<<grid, 256, 0, stream>>>(phi, sens, gamma, alpha, mixw,
                                              out, partials);
    ift_reduce_kernel<<<10, 256, 0, stream>>>(partials, out);
}